// StdWeightedLoss_8985071583756
// MI455X (gfx1250) — compile-verified
//
#include <hip/hip_runtime.h>
#include <math.h>

// StdWeightedLoss for MI455X (gfx1250, wave32).
// Memory-bound: 2 x 100.7MB fp32 reads -> ~9us floor @ 23.3 TB/s.
// Separable 11x11 box filter in LDS; fp32 WMMA (16x16x4_f32) for the
// deterministic intra-block reductions; target tile staged into LDS with
// CDNA5 async global->LDS copies (ASYNCcnt), predict warmed via
// global_prefetch while copies are in flight.

typedef __attribute__((ext_vector_type(2))) float v2f;
typedef __attribute__((ext_vector_type(8))) float v8f;

#define KS    11
#define PADR  5
#define TW    64
#define TH    32
#define PH    (TH + KS - 1)   // 42 padded rows
#define PW    (TW + KS - 1)   // 74 padded cols
#define PWS   76              // LDS row stride for raw tile
#define HH    512
#define WW    512
#define NTHREADS 256

// Reduce 256 fp32 values (in LDS) with V_WMMA_F32_16X16X4_F32.
// A = 16x4 chunk of data (2 VGPRs/lane), B = all-ones 4x16, C accumulates.
// With B == 1, every column of D holds the 16 row-sums of the accumulated A
// chunks, independent of the exact A lane striping (any bijection of 64
// values onto A slots works). Lanes 0-15 hold rows M=0..7, lanes 16-31 hold
// rows M=8..15, so total = sum(regs) + shfl_xor(sum(regs), 16).
// Must be called with the full wave active (EXEC all ones).
__device__ __forceinline__ float wmma_reduce256(const float* red, int lane) {
  v8f acc = {};
  v2f ones;
  ones.x = 1.0f;
  ones.y = 1.0f;
#pragma unroll
  for (int chunk = 0; chunk < 4; ++chunk) {
    v2f a;
    a.x = red[chunk * 64 + 2 * lane];
    a.y = red[chunk * 64 + 2 * lane + 1];
    acc = __builtin_amdgcn_wmma_f32_16x16x4_f32(
        /*neg_a=*/false, a, /*neg_b=*/false, ones,
        /*c_mod=*/(short)0, acc, /*reuse_a=*/false, /*reuse_b=*/false);
  }
  float s = 0.0f;
#pragma unroll
  for (int r = 0; r < 8; ++r) s += acc[r];
  s += __shfl_xor(s, 16, 32);   // combine M=0..7 half with M=8..15 half
  return s;
}

__global__ void __launch_bounds__(NTHREADS)
std_loss_tile_kernel(const float* __restrict__ predict,
                     const float* __restrict__ target,
                     float* __restrict__ partials) {
  __shared__ float t_raw[PH * PWS];   // 12768 B
  __shared__ float hsum[PH * TW];     // 10752 B
  __shared__ float hsq[PH * TW];      // 10752 B
  __shared__ float red[NTHREADS];     //  1024 B  (~35 KB total, 320 KB/WGP)

  const int tid = threadIdx.x;
  const int tile_c0 = blockIdx.x * TW;
  const int tile_r0 = blockIdx.y * TH;
  const size_t plane_off = (size_t)blockIdx.z * (size_t)(HH * WW);
  const float* __restrict__ tp = target + plane_off;
  const float* __restrict__ pp = predict + plane_off;

  // ---- replicate-padded target tile -> LDS via async global->LDS copies ----
  // Per-lane clamped global address (VADDR) + per-lane LDS byte offset (VDST);
  // no VGPR staging, tracked on ASYNCcnt so it overlaps the prefetch below.
  for (int i = tid; i < PH * PW; i += NTHREADS) {
    int r = i / PW;
    int c = i - r * PW;
    int gr = tile_r0 + r - PADR;
    gr = gr < 0 ? 0 : (gr > HH - 1 ? HH - 1 : gr);
    int gc = tile_c0 + c - PADR;
    gc = gc < 0 ? 0 : (gc > WW - 1 ? WW - 1 : gc);
    // Low 32 bits of a flat pointer into LDS == offset in the WG allocation.
    unsigned lds_off = (unsigned)(size_t)&t_raw[r * PWS + c];
    unsigned long long gaddr =
        (unsigned long long)(size_t)(tp + ((size_t)gr * WW + gc));
    asm volatile("global_load_async_to_lds_b32 %0, %1, off"
                 :: "v"(lds_off), "v"(gaddr)
                 : "memory");
  }

  // Warm L2 with the predict tile while the async copies are in flight
  // (global_prefetch_b8; one byte/lane pulls whole cachelines).
  {
    int pr = tid >> 3;            // 0..31 rows
    int pc = (tid & 7) * 8;       // 8 probes across 64 cols
    __builtin_prefetch(&pp[(size_t)(tile_r0 + pr) * WW + tile_c0 + pc], 0, 1);
  }

  // Drain this wave's async copies, then make LDS visible workgroup-wide.
  asm volatile("s_wait_asynccnt 0x0" ::: "memory");
  __syncthreads();

  // ---- horizontal 11-tap sums of t and t^2 ----
  for (int i = tid; i < PH * TW; i += NTHREADS) {
    int r = i >> 6;               // /TW
    int c = i & (TW - 1);
    const float* row = &t_raw[r * PWS + c];
    float s = 0.0f, sq = 0.0f;
#pragma unroll
    for (int j = 0; j < KS; ++j) {
      float v = row[j];
      s += v;
      sq = fmaf(v, v, sq);
    }
    hsum[i] = s;
    hsq[i] = sq;
  }
  __syncthreads();

  // ---- vertical 11-tap + weight + weighted-L1 accumulation ----
  const float inv = 1.0f / (float)(KS * KS);
  float acc = 0.0f;
  for (int i = tid; i < TH * TW; i += NTHREADS) {   // 8 pixels/thread
    int r = i >> 6;
    int c = i & (TW - 1);
    float s = 0.0f, sq = 0.0f;
#pragma unroll
    for (int j = 0; j < KS; ++j) {
      s += hsum[(r + j) * TW + c];
      sq += hsq[(r + j) * TW + c];
    }
    float mean = s * inv;
    float var = fmaf(-mean, mean, sq * inv);
    var = var > 0.0f ? var : 0.0f;
    float sd = sqrtf(var);
    sd = sd < 0.4f ? sd : 0.4f;
    float w = fmaf(sd, 5.0f, 1.0f);                 // 1 + min(std,0.4)/0.2
    float tv = t_raw[(r + PADR) * PWS + (c + PADR)];
    float pv = pp[(size_t)(tile_r0 + r) * WW + (tile_c0 + c)];
    acc = fmaf(fabsf(pv - tv), w, acc);
  }

  red[tid] = acc;
  __syncthreads();

  if (tid < 32) {                                   // wave 0 fully active
    float total = wmma_reduce256(red, tid);
    if (tid == 0) {
      partials[((size_t)blockIdx.z * gridDim.y + blockIdx.y) * gridDim.x +
               blockIdx.x] = total;
    }
  }
}

__global__ void __launch_bounds__(NTHREADS)
std_loss_final_kernel(const float* __restrict__ partials, int n,
                      float* __restrict__ out, float inv_count) {
  __shared__ float red[NTHREADS];
  const int tid = threadIdx.x;
  float s = 0.0f;
  // fixed-order strided fold -> deterministic
  for (int i = tid; i < n; i += NTHREADS) s += partials[i];
  red[tid] = s;
  __syncthreads();
  if (tid < 32) {
    float total = wmma_reduce256(red, tid);
    if (tid == 0) out[0] = total * inv_count;
  }
}

extern "C" void kernel_launch(void* const* d_in, const int* in_sizes, int n_in,
                              void* d_out, int out_size, void* d_ws, size_t ws_size,
                              hipStream_t stream) {
  const float* predict = (const float*)d_in[0];
  const float* target  = (const float*)d_in[1];
  float* out = (float*)d_out;
  float* partials = (float*)d_ws;

  const int total_elems = in_sizes[0];              // 32*3*512*512
  const int planes = total_elems / (HH * WW);       // 96
  dim3 grid(WW / TW, HH / TH, planes);              // 8 x 16 x 96 = 12288
  dim3 block(NTHREADS);

  std_loss_tile_kernel<<<grid, block, 0, stream>>>(predict, target, partials);

  const int n_partials = (int)(grid.x * grid.y * grid.z);
  const float inv_count = 1.0f / (float)total_elems;
  std_loss_final_kernel<<<1, block, 0, stream>>>(partials, n_partials, out,
                                                 inv_count);
}